// SSDIntNBitTableBatchedEmbeddingBags_55310588838091
// MI455X (gfx1250) — compile-verified
//
#include <hip/hip_runtime.h>

// Problem constants (from the reference): T tables, E rows, D dim, B bags, L pooling.
#define T_ 4
#define E_ 500000
#define D_ 128
#define B_ 4096
#define L_ 50

#define WAVES   8    // waves per block (wave32)
#define SLOTS   8    // LDS row slots per wave, 512B each -> 4KB/wave, 32KB/block
#define IDXW    16   // rolling index-register window
#define PF_ROW  7    // async row prefetch distance
#define PF_IDX  14   // index prefetch distance

typedef _Float16 half4_t __attribute__((ext_vector_type(4)));

__global__ __launch_bounds__(256) void
embed_pool_int8_kernel(const int*   __restrict__ indices,   // [T,B,L]
                       const int*   __restrict__ qweights,  // [T,E,D] byte values in int32
                       const float* __restrict__ scales,    // [T,E]
                       const float* __restrict__ biases,    // [T,E]
                       _Float16*    __restrict__ out)       // [B, T*D] fp16
{
    __shared__ __align__(16) int buf[WAVES][SLOTS][D_];

    const int lane = threadIdx.x & 31;
    const int wid  = threadIdx.x >> 5;
    const int w    = blockIdx.x * WAVES + wid;   // global wave id == (t,b)
    const int t    = w >> 12;                    // / B_ (B_ = 4096)
    const int b    = w & (B_ - 1);

    const int*   idxp = indices + ((size_t)t * B_ + b) * L_;
    const float* scp  = scales + (size_t)t * E_;
    const float* bip  = biases + (size_t)t * E_;
    // 64-bit per-lane global base: table base + this lane's 16B chunk.
    const unsigned long long qbase =
        (unsigned long long)(size_t)(qweights + (size_t)t * E_ * D_)
        + (unsigned long long)((unsigned)lane << 4);

    // Low 32 bits of a generic shared-space address are the raw LDS byte
    // offset -- what the async
    // instruction's VDST operand wants.
    const unsigned lbase =
        (unsigned)(size_t)(&buf[wid][0][0]) + ((unsigned)lane << 4);

    int   idxw[IDXW];
    float scw[SLOTS];
    float biw[SLOTS];

    float a0 = 0.f, a1 = 0.f, a2 = 0.f, a3 = 0.f, bsum = 0.f;

    // Phase 0: prime the index window (indices are contiguous per bag).
#pragma unroll
    for (int l = 0; l < PF_IDX; ++l) idxw[l] = idxp[l];

    // Phase 1: launch the first PF_ROW async row gathers + scale/bias loads.
#pragma unroll
    for (int l = 0; l < PF_ROW; ++l) {
        const int idx = idxw[l & (IDXW - 1)];
        scw[l & (SLOTS - 1)] = scp[idx];
        biw[l & (SLOTS - 1)] = bip[idx];
        const unsigned ldsoff = lbase + (unsigned)((l & (SLOTS - 1)) * (D_ * 4));
        const unsigned long long gaddr =
            qbase + (unsigned long long)((unsigned)idx * (unsigned)(D_ * 4));
        asm volatile("global_load_async_to_lds_b128 %0, %1, off offset:0"
                     :: "v"(ldsoff), "v"(gaddr) : "memory");
    }

    // Phase 2: steady-state software pipeline (fully unrolled, L_=50).
#pragma unroll
    for (int l = 0; l < L_; ++l) {
        // Prefetch a future index into the rolling window.
        if (l + PF_IDX < L_) idxw[(l + PF_IDX) & (IDXW - 1)] = idxp[l + PF_IDX];

        if (l + PF_ROW < L_) {
            // The slot we are about to overwrite was consumed last
            // iteration; async-LDS writes are NOT ordered against pending
            // ds reads from this wave, so drain DScnt first (cheap: those
            // reads retired when their values were used).
            asm volatile("s_wait_dscnt 0" ::: "memory");
            const int li  = l + PF_ROW;
            const int idx = idxw[li & (IDXW - 1)];
            scw[li & (SLOTS - 1)] = scp[idx];
            biw[li & (SLOTS - 1)] = bip[idx];
            const unsigned ldsoff = lbase + (unsigned)((li & (SLOTS - 1)) * (D_ * 4));
            const unsigned long long gaddr =
                qbase + (unsigned long long)((unsigned)idx * (unsigned)(D_ * 4));
            asm volatile("global_load_async_to_lds_b128 %0, %1, off offset:0"
                         :: "v"(ldsoff), "v"(gaddr) : "memory");
        }

        // Async loads retire in order: waiting ASYNCcnt<=N pops exactly row l.
        asm volatile("s_wait_asynccnt %0"
                     :: "n"((L_ - 1 - l) < PF_ROW ? (L_ - 1 - l) : PF_ROW)
                     : "memory");

        // Consume row l from LDS: ds_load_b128, dequant + SUM-pool.
        const int4  q = *(const int4*)(&buf[wid][l & (SLOTS - 1)][lane << 2]);
        const float s = scw[l & (SLOTS - 1)];
        a0 = fmaf((float)q.x, s, a0);
        a1 = fmaf((float)q.y, s, a1);
        a2 = fmaf((float)q.z, s, a2);
        a3 = fmaf((float)q.w, s, a3);
        bsum += biw[l & (SLOTS - 1)];
    }

    half4_t o;
    o.x = (_Float16)(a0 + bsum);
    o.y = (_Float16)(a1 + bsum);
    o.z = (_Float16)(a2 + bsum);
    o.w = (_Float16)(a3 + bsum);
    *(half4_t*)(out + (size_t)b * (T_ * D_) + (size_t)t * D_ + (lane << 2)) = o;
}

extern "C" void kernel_launch(void* const* d_in, const int* in_sizes, int n_in,
                              void* d_out, int out_size, void* d_ws, size_t ws_size,
                              hipStream_t stream) {
    const int*   indices  = (const int*)  d_in[0];
    const int*   qweights = (const int*)  d_in[1];
    const float* scales   = (const float*)d_in[2];
    const float* biases   = (const float*)d_in[3];
    _Float16*    out      = (_Float16*)   d_out;

    dim3 grid((T_ * B_) / WAVES);   // 16384 waves, one per (table, bag)
    dim3 block(WAVES * 32);         // 256 threads = 8 wave32
    hipLaunchKernelGGL(embed_pool_int8_kernel, grid, block, 0, stream,
                       indices, qweights, scales, biases, out);
}